// SphereSampler_3375844295211
// MI455X (gfx1250) — compile-verified
//
#include <hip/hip_runtime.h>
#include <hip/hip_bf16.h>
#include <stdint.h>

// Volume geometry (matches reference)
#define DD 128
#define HH 128
#define WW 128
#define NSPH 64
#define VOXELS (DD * HH * WW)        // 2,097,152
#define SLICE  2097152               // elements per sphere in noise
#define VPT 4                        // voxels (x-consecutive) per thread

// LDS parameter block: centers[192] | radii[64] | labels[64] = 320 dwords
#define P_CEN 0
#define P_RAD 192
#define P_LAB 256
#define P_TOT 320

#if __has_builtin(__builtin_amdgcn_tensor_load_to_lds) && __has_builtin(__builtin_amdgcn_s_wait_tensorcnt)
#define USE_TDM 1
#else
#define USE_TDM 0
#endif

#if USE_TDM
typedef unsigned int u32x4 __attribute__((ext_vector_type(4)));
typedef int          i32x8 __attribute__((ext_vector_type(8)));
typedef int          i32x4 __attribute__((ext_vector_type(4)));

// Issue one 1-D TDM copy: ndw dwords from global gptr -> LDS byte offset lds_off.
// Descriptor layout per CDNA5 ISA ch.8 (D# groups 0/1; trailing groups zero => <=2D).
__device__ __forceinline__ void tdm_copy_1d(uint32_t lds_off, const void* gptr,
                                            uint32_t ndw) {
  uint64_t ga = (uint64_t)(uintptr_t)gptr;
  u32x4 g0 = {
      1u,                                           // count=1, user mode
      lds_off,                                      // lds_addr (bytes)
      (uint32_t)ga,                                 // global_addr[31:0]
      (uint32_t)((ga >> 32) & 0x01FFFFFFu) | (2u << 30)  // addr[56:32] | type=2
  };
  i32x8 g1 = {
      (int)(2u << 16),                              // workgroup_mask=0, data_size=4B
      (int)((ndw & 0xFFFFu) << 16),                 // tensor_dim0[15:0]
      (int)((ndw >> 16) | (1u << 16)),              // tensor_dim0[31:16], tensor_dim1=1
      (int)(ndw << 16),                             // tile_dim0 = ndw
      0,                                            // tile_dim1=0 (unused), tile_dim2=0
      (int)ndw,                                     // tensor_dim0_stride lo
      0, 0                                          // stride hi, tensor_dim1_stride
  };
  i32x4 z4 = {0, 0, 0, 0};
  i32x8 z8 = {0, 0, 0, 0, 0, 0, 0, 0};
  __builtin_amdgcn_tensor_load_to_lds(g0, g1, z4, z4, z8, 0);
}
#endif

__global__ __launch_bounds__(256) void sphere_sampler_kernel(
    const float* __restrict__ noise,     // [64][128][128][128]
    const float* __restrict__ labels,    // [64]
    const int*   __restrict__ centers,   // [64][3]
    const int*   __restrict__ radii,     // [64]
    float*       __restrict__ out) {     // [128][128][128]

  __shared__ int sP[P_TOT];

#if USE_TDM
  if (threadIdx.x == 0) {
    // Low 32 bits of a generic LDS pointer = LDS byte offset on AMDGPU.
    uint32_t base = (uint32_t)(uintptr_t)(void*)&sP[0];
    tdm_copy_1d(base + 4u * P_CEN, centers, 192u);
    tdm_copy_1d(base + 4u * P_RAD, radii, 64u);
    tdm_copy_1d(base + 4u * P_LAB, labels, 64u);
    __builtin_amdgcn_s_wait_tensorcnt(0);
  }
#else
  for (int t = threadIdx.x; t < P_TOT; t += blockDim.x) {
    int v;
    if (t < P_RAD)      v = centers[t];
    else if (t < P_LAB) v = radii[t - P_RAD];
    else                v = __float_as_int(labels[t - P_LAB]);
    sP[t] = v;
  }
#endif
  __syncthreads();

  // 4 consecutive-x voxels per thread: same (z,y) row, 16B-aligned segment.
  int idx = (int)(blockIdx.x * blockDim.x + threadIdx.x) * VPT;
  if (idx >= VOXELS) return;
  int x0 = idx & (WW - 1);
  int y  = (idx >> 7) & (HH - 1);
  int z  = idx >> 14;

  float val[VPT] = {0.0f, 0.0f, 0.0f, 0.0f};
  const float* np = noise + idx;

  #pragma unroll 4
  for (int s = 0; s < NSPH; ++s) {
    int cz = sP[P_CEN + 3 * s + 0];
    int cy = sP[P_CEN + 3 * s + 1];
    int cx = sP[P_CEN + 3 * s + 2];
    int r  = sP[P_RAD + s];
    int dz = z - cz, dy = y - cy, dx0 = x0 - cx;
    float base = (float)(dz * dz + dy * dy);   // shared across the 4 voxels
    float d2[VPT];
    #pragma unroll
    for (int i = 0; i < VPT; ++i) {
      int t = dx0 + i;
      d2[i] = base + (float)(t * t);
    }
    // rad = r + 0.5*N(0,1); beyond r+6 (12 sigma) a voxel can never be inside,
    // so skip the noise load when the whole 4-voxel segment is out of band
    // (cuts 512MB of HBM noise traffic to ~10MB).
    float rb  = (float)r + 6.0f;
    float rb2 = rb * rb;
    bool any = (d2[0] <= rb2) | (d2[1] <= rb2) | (d2[2] <= rb2) | (d2[3] <= rb2);
    if (any) {
      const float4 n4 = *(const float4*)(np + (long)s * SLICE);  // b128 load
      float nn[VPT] = {n4.x, n4.y, n4.z, n4.w};
      float std = (r < 3) ? 0.001f : 0.5f;     // faithful to reference select
      float lab = __int_as_float(sP[P_LAB + s]);
      #pragma unroll
      for (int i = 0; i < VPT; ++i) {
        float rad = (float)r + nn[i] * std;
        if (d2[i] <= rad * rad) val[i] = lab;
      }
    }
  }
  float4 o = {val[0], val[1], val[2], val[3]};
  *(float4*)(out + idx) = o;                    // b128 store
}

extern "C" void kernel_launch(void* const* d_in, const int* in_sizes, int n_in,
                              void* d_out, int out_size, void* d_ws, size_t ws_size,
                              hipStream_t stream) {
  const float* noise   = (const float*)d_in[0];
  const float* labels  = (const float*)d_in[1];
  const int*   centers = (const int*)d_in[2];
  const int*   radii   = (const int*)d_in[3];
  float*       out     = (float*)d_out;

  dim3 block(256);
  dim3 grid(VOXELS / (256 * VPT));   // 2048 blocks
  sphere_sampler_kernel<<<grid, block, 0, stream>>>(noise, labels, centers, radii, out);
}